// TrajectoryInference_75118978007310
// MI455X (gfx1250) — compile-verified
//
#include <hip/hip_runtime.h>

#define EMBED_DIM 256
#define D1 128
#define D2 64
#define NCOMP 50
#define KPAD 64
#define BB 4
#define NN 4096

typedef __attribute__((ext_vector_type(2))) float v2f;
typedef __attribute__((ext_vector_type(4))) float v4f;
typedef __attribute__((ext_vector_type(8))) float v8f;

// Swizzled position of logical k (0..63) within a stored 64-float z row.
// Groups the WMMA K-pairs so each lane-half's data is 32 contiguous floats:
//   half 0 (lanes 0-15):  k === {0,1} mod 4  -> positions [0,32)
//   half 1 (lanes 16-31): k === {2,3} mod 4  -> positions [32,64)
// Within a half, step s = k>>2 occupies positions {2s, 2s+1}.
__device__ __forceinline__ int zswz(int k) {
  return ((k & 2) >> 1) * 32 + (k >> 2) * 2 + (k & 1);
}

// ---------------------------------------------------------------------------
// Kernel A: MLP 256 -> 128 -> 64 -> 50 per row; writes z in the WMMA-swizzled
// KPAD=64 layout (zero padded) plus sq[i] = z.z. One 256-thread block per row.
// ---------------------------------------------------------------------------
__global__ __launch_bounds__(256) void mlp_kernel(
    const float* __restrict__ emb,
    const float* __restrict__ W1, const float* __restrict__ b1,
    const float* __restrict__ W2, const float* __restrict__ b2,
    const float* __restrict__ W3, const float* __restrict__ b3,
    float* __restrict__ z64, float* __restrict__ sq) {
  __shared__ float sE[EMBED_DIM];
  __shared__ float sH1[D1];
  __shared__ float sH2[D2];
  __shared__ float sZ2[KPAD];

  const int row = blockIdx.x;           // 0 .. B*N-1
  const int t = threadIdx.x;            // 0 .. 255

  sE[t] = emb[(size_t)row * EMBED_DIM + t];
  __syncthreads();

  if (t < D1) {
    float acc = b1[t];
    #pragma unroll 8
    for (int d = 0; d < EMBED_DIM; ++d) acc += sE[d] * W1[d * D1 + t];
    sH1[t] = fmaxf(acc, 0.0f);
  }
  __syncthreads();

  if (t < D2) {
    float acc = b2[t];
    #pragma unroll 8
    for (int d = 0; d < D1; ++d) acc += sH1[d] * W2[d * D2 + t];
    sH2[t] = fmaxf(acc, 0.0f);
  }
  __syncthreads();

  if (t < KPAD) {
    float zv = 0.0f;
    if (t < NCOMP) {
      float acc = b3[t];
      #pragma unroll 8
      for (int d = 0; d < D2; ++d) acc += sH2[d] * W3[d * NCOMP + t];
      zv = acc;
    }
    z64[(size_t)row * KPAD + zswz(t)] = zv;   // swizzled store
    sZ2[t] = zv * zv;
  }
  __syncthreads();

  if (t == 0) {
    float s = 0.0f;
    #pragma unroll
    for (int d = 0; d < KPAD; ++d) s += sZ2[d];
    sq[row] = s;
  }
}

// ---------------------------------------------------------------------------
// One wave computes a 16x16 gram tile via V_WMMA_F32_16X16X4_F32.
// With the swizzled layout each lane issues 8 independent b128 loads per
// operand (one clause, one wait), then 16 back-to-back WMMAs.
// ---------------------------------------------------------------------------
__device__ __forceinline__ v8f gram_tile(const float* __restrict__ zb,
                                         int i0, int j0, int lane) {
  const int r = lane & 15;
  const int halfOff = (lane >> 4) << 5;   // 0 or 32 floats
  const v4f* __restrict__ Ai = (const v4f*)(zb + (size_t)(i0 + r) * KPAD + halfOff);
  const v4f* __restrict__ Bj = (const v4f*)(zb + (size_t)(j0 + r) * KPAD + halfOff);

  v4f a[8], b[8];
  #pragma unroll
  for (int q = 0; q < 8; ++q) { a[q] = Ai[q]; b[q] = Bj[q]; }

  v8f c = {0.f, 0.f, 0.f, 0.f, 0.f, 0.f, 0.f, 0.f};
  #pragma unroll
  for (int q = 0; q < 8; ++q) {
    c = __builtin_amdgcn_wmma_f32_16x16x4_f32(false, a[q].lo, false, b[q].lo,
                                              (short)0, c, false, false);
    c = __builtin_amdgcn_wmma_f32_16x16x4_f32(false, a[q].hi, false, b[q].hi,
                                              (short)0, c, false, false);
  }
  return c;
}

// t_ij = exp(exp(-max(dist,0)/2) - 1); the diagonal kernel value is exactly 1
// (the row max), so the softmax needs no max-scan pass.
__device__ __forceinline__ float tile_t(float sqi, float sqj, float g) {
  float dist = fmaxf(sqi + sqj - 2.0f * g, 0.0f);
  float k = __expf(-0.5f * dist);
  return __expf(k - 1.0f);
}

// Prefetch next chunk's half-row for this lane (lowers to global_prefetch_b8).
__device__ __forceinline__ void prefetch_next(const float* __restrict__ zb,
                                              int chunk, int wave, int lane) {
  const int jn = ((chunk + 128) & (NN - 1)) + (wave << 4) + (lane & 15);
  const float* pf = zb + (size_t)jn * KPAD + ((lane >> 4) << 5);
  __builtin_prefetch(pf, 0, 0);
}

// ---------------------------------------------------------------------------
// Kernel B: row sums S_i = sum_j exp(k_ij - 1).
// Block = 8 waves owns 16 rows x all 4096 cols; each wave takes a 16-col
// slice per 128-col chunk. Grid = B * N/16 = 1024 blocks.
// ---------------------------------------------------------------------------
__global__ __launch_bounds__(256) void rowsum_kernel(
    const float* __restrict__ z64, const float* __restrict__ sq,
    float* __restrict__ S) {
  const int b = blockIdx.x >> 8;
  const int i0 = (blockIdx.x & 255) << 4;
  const float* __restrict__ zb = z64 + (size_t)b * NN * KPAD;
  const float* __restrict__ sqb = sq + (size_t)b * NN;

  const int lane = threadIdx.x & 31;
  const int wave = threadIdx.x >> 5;
  const int hi = (lane >> 4) << 3;            // C/D row-group offset: 0 or 8

  float sqi[8];
  #pragma unroll
  for (int e = 0; e < 8; ++e) sqi[e] = sqb[i0 + hi + e];

  __shared__ float Ssum[16];
  if (threadIdx.x < 16) Ssum[threadIdx.x] = 0.0f;
  __syncthreads();

  float acc[8] = {0.f, 0.f, 0.f, 0.f, 0.f, 0.f, 0.f, 0.f};
  for (int chunk = 0; chunk < NN; chunk += 128) {
    const int j0 = chunk + (wave << 4);
    prefetch_next(zb, chunk, wave, lane);
    v8f c = gram_tile(zb, i0, j0, lane);
    const float sqj = sqb[j0 + (lane & 15)];
    #pragma unroll
    for (int e = 0; e < 8; ++e) acc[e] += tile_t(sqi[e], sqj, c[e]);
  }

  #pragma unroll
  for (int e = 0; e < 8; ++e) atomicAdd(&Ssum[hi + e], acc[e]);
  __syncthreads();
  if (threadIdx.x < 16) S[(size_t)b * NN + i0 + threadIdx.x] = Ssum[threadIdx.x];
}

// ---------------------------------------------------------------------------
// Kernel C: recompute tiles (~7 GFLOP, free) and stream the 268 MB output:
// out = (t/S_i + 1e-6) / (1 + N*1e-6)   [softmax rows sum to exactly 1]
// ---------------------------------------------------------------------------
__global__ __launch_bounds__(256) void out_kernel(
    const float* __restrict__ z64, const float* __restrict__ sq,
    const float* __restrict__ S, float* __restrict__ out) {
  const int b = blockIdx.x >> 8;
  const int i0 = (blockIdx.x & 255) << 4;
  const float* __restrict__ zb = z64 + (size_t)b * NN * KPAD;
  const float* __restrict__ sqb = sq + (size_t)b * NN;
  const float* __restrict__ Sb = S + (size_t)b * NN;
  float* __restrict__ ob = out + (size_t)b * NN * NN;

  const int lane = threadIdx.x & 31;
  const int wave = threadIdx.x >> 5;
  const int hi = (lane >> 4) << 3;

  const float renorm = 1.0f / (1.0f + (float)NN * 1e-6f);

  float sqi[8], invS[8];
  #pragma unroll
  for (int e = 0; e < 8; ++e) {
    sqi[e] = sqb[i0 + hi + e];
    invS[e] = 1.0f / Sb[i0 + hi + e];
  }

  for (int chunk = 0; chunk < NN; chunk += 128) {
    const int j0 = chunk + (wave << 4);
    prefetch_next(zb, chunk, wave, lane);
    v8f c = gram_tile(zb, i0, j0, lane);
    const float sqj = sqb[j0 + (lane & 15)];
    #pragma unroll
    for (int e = 0; e < 8; ++e) {
      float t = tile_t(sqi[e], sqj, c[e]);
      float v = (t * invS[e] + 1e-6f) * renorm;
      ob[(size_t)(i0 + hi + e) * NN + j0 + (lane & 15)] = v;
    }
  }
}

// ---------------------------------------------------------------------------
extern "C" void kernel_launch(void* const* d_in, const int* in_sizes, int n_in,
                              void* d_out, int out_size, void* d_ws, size_t ws_size,
                              hipStream_t stream) {
  const float* emb = (const float*)d_in[0];
  const float* W1  = (const float*)d_in[1];
  const float* b1  = (const float*)d_in[2];
  const float* W2  = (const float*)d_in[3];
  const float* b2  = (const float*)d_in[4];
  const float* W3  = (const float*)d_in[5];
  const float* b3  = (const float*)d_in[6];

  float* z64 = (float*)d_ws;                         // B*N*64 floats = 4 MB
  float* sqv = z64 + (size_t)BB * NN * KPAD;         // B*N floats
  float* Sv  = sqv + (size_t)BB * NN;                // B*N floats
  float* out = (float*)d_out;                        // (B, N, N) f32

  mlp_kernel<<<BB * NN, 256, 0, stream>>>(emb, W1, b1, W2, b2, W3, b3, z64, sqv);
  rowsum_kernel<<<BB * (NN / 16), 256, 0, stream>>>(z64, sqv, Sv);
  out_kernel<<<BB * (NN / 16), 256, 0, stream>>>(z64, sqv, Sv, out);
}